// ChamferLoss_34351148433808
// MI455X (gfx1250) — compile-verified
//
#include <hip/hip_runtime.h>
#include <hip/hip_bf16.h>

typedef float v2f __attribute__((ext_vector_type(2)));
typedef float v8f __attribute__((ext_vector_type(8)));

#define TILE 16
#define NPTS 4096

// Zero the scalar accumulator (harness poisons d_out with 0xAA).
__global__ void chamfer_zero_kernel(float* out) {
    if (threadIdx.x == 0 && blockIdx.x == 0) out[0] = 0.0f;
}

// One wave per (batch, 16-row tile of P). For each 16-col tile of T, one
// V_WMMA_F32_16X16X4_F32 produces the 16x16 squared-distance tile:
//   A[m] = (-2px, -2py, -2pz, 1)   (16x4, K padded 3->4)
//   B[n] = ( tx,   ty,   tz, ||t||^2)  (4x16)
//   C[m][n] = ||p_m||^2
//   D = A*B + C = ||p||^2 + ||t||^2 - 2 p.t
// Row mins are kept in the 8 accumulator-layout VGPRs, lane-reduced, and the
// wave atomically adds its (pre-scaled) partial sum of row-mins to out[0].
// Called twice with (P,T) and (T,P) to get both Chamfer directions.
__global__ void __launch_bounds__(256)
chamfer_dir_kernel(const float* __restrict__ P, const float* __restrict__ T,
                   float* __restrict__ out, int B, float scale) {
    const int lane = threadIdx.x & 31;
    const int wave = (blockIdx.x * blockDim.x + threadIdx.x) >> 5;
    const int tilesPerBatch = NPTS / TILE;           // 256
    const int b  = wave / tilesPerBatch;
    const int rt = wave % tilesPerBatch;             // row tile in P
    if (b >= B) return;                              // never taken (grid exact)

    const int sub = lane & 15;                       // col within tile / row M
    const int hi  = lane >> 4;                       // upper-lane half flag

    // ---- A matrix (16x4 f32): low lanes hold K=0,1; high lanes K=2,3 ----
    const size_t pbase = ((size_t)b * NPTS + (size_t)rt * TILE) * 3;
    {
        // nothing
    }
    const float* pm = P + pbase + (size_t)sub * 3;
    float px = pm[0], py = pm[1], pz = pm[2];
    v2f a;
    a.x = hi ? (-2.0f * pz) : (-2.0f * px);          // K=2 : K=0
    a.y = hi ? 1.0f         : (-2.0f * py);          // K=3 : K=1

    // ---- C matrix (16x16 f32): VGPR r -> M = r + 8*hi, value ||p_M||^2 ----
    v8f c;
    const float* prow = P + pbase + (size_t)(8 * hi) * 3;
    #pragma unroll
    for (int r = 0; r < 8; ++r) {
        float x = prow[3 * r + 0];
        float y = prow[3 * r + 1];
        float z = prow[3 * r + 2];
        c[r] = x * x + y * y + z * z;
    }

    // ---- sweep all column tiles of T ----
    v8f rmin;
    #pragma unroll
    for (int r = 0; r < 8; ++r) rmin[r] = 3.0e38f;

    const float* tb = T + (size_t)b * NPTS * 3;
    for (int ct = 0; ct < tilesPerBatch; ++ct) {
        const float* tn = tb + (size_t)(ct * TILE + sub) * 3;
        float tx = tn[0], ty = tn[1], tz = tn[2];
        float t2 = tx * tx + ty * ty + tz * tz;
        v2f bb;
        bb.x = hi ? tz : tx;                         // K=2 : K=0
        bb.y = hi ? t2 : ty;                         // K=3 : K=1

        v8f d = __builtin_amdgcn_wmma_f32_16x16x4_f32(
            /*neg_a=*/false, a, /*neg_b=*/false, bb,
            /*c_mod=*/(short)0, c, /*reuse_a=*/false, /*reuse_b=*/false);

        #pragma unroll
        for (int r = 0; r < 8; ++r) rmin[r] = fminf(rmin[r], d[r]);
    }

    // ---- reduce row-mins across the 16 lanes of each half (cols) ----
    float s = 0.0f;
    #pragma unroll
    for (int r = 0; r < 8; ++r) {
        float v = rmin[r];
        v = fminf(v, __shfl_xor(v, 1, 32));
        v = fminf(v, __shfl_xor(v, 2, 32));
        v = fminf(v, __shfl_xor(v, 4, 32));
        v = fminf(v, __shfl_xor(v, 8, 32));
        s += v;                                      // low half: M=r, high: M=r+8
    }
    s += __shfl_xor(s, 16, 32);                      // combine both M halves
    if (lane == 0) atomicAdd(out, s * scale);
}

extern "C" void kernel_launch(void* const* d_in, const int* in_sizes, int n_in,
                              void* d_out, int out_size, void* d_ws, size_t ws_size,
                              hipStream_t stream) {
    const float* pred   = (const float*)d_in[0];
    const float* target = (const float*)d_in[1];
    float* out = (float*)d_out;

    const int B = in_sizes[0] / (NPTS * 3);          // 8
    const float scale = 1.0f / ((float)B * (float)NPTS);

    chamfer_zero_kernel<<<1, 32, 0, stream>>>(out);

    const int wavesPerDir = B * (NPTS / TILE);       // 2048 waves
    const int block = 256;                           // 8 waves/block
    const int grid  = wavesPerDir * 32 / block;      // 256 blocks

    // Direction 1: for each pred row, min over targets.
    chamfer_dir_kernel<<<grid, block, 0, stream>>>(pred, target, out, B, scale);
    // Direction 2: for each target row, min over preds.
    chamfer_dir_kernel<<<grid, block, 0, stream>>>(target, pred, out, B, scale);
}